// PreNorm_31731218382919
// MI455X (gfx1250) — compile-verified
//
#include <hip/hip_runtime.h>
#include <cstddef>

// ---------------------------------------------------------------------------
// Fused PreNorm + QKV projection + KV-cache-update attention for MI455X
// (gfx1250, wave32, WMMA). fp32 end-to-end via V_WMMA_F32_16X16X4_F32.
//
// Pipeline (4 kernels on one stream):
//   1) rms     : per-row inv RMS of X                      [32]
//   2) qkv     : (X*invr) @ W_qkv  -> ws.qkv [32 x 6144]   (WMMA)
//   3) attn    : per (head, L-split) streaming softmax-attention partials.
//                K chunks double-buffered in LDS via async global->LDS b128
//                (ASYNCcnt pipelined), V streamed from HBM. Rows [P,P+32)
//                come from the fresh k/v in ws.qkv — inputs never mutated.
//   4) reduce  : combine partials -> out [32 x 2048]
// ---------------------------------------------------------------------------

typedef float v2f __attribute__((ext_vector_type(2)));
typedef float v8f __attribute__((ext_vector_type(8)));
typedef int   v4i __attribute__((vector_size(4 * sizeof(int))));

#if defined(__has_builtin)
#if __has_builtin(__builtin_amdgcn_global_load_async_to_lds_b128)
#define USE_ASYNC_LDS 1
#endif
#endif
#ifndef USE_ASYNC_LDS
#define USE_ASYNC_LDS 0
#endif

#if USE_ASYNC_LDS
#if __has_builtin(__builtin_amdgcn_s_wait_asynccnt)
#define WAIT_ASYNC(n) __builtin_amdgcn_s_wait_asynccnt(n)
#else
#define WAIT_ASYNC(n) asm volatile("s_wait_asynccnt %0" ::"n"(n) : "memory")
#endif
#else
#define WAIT_ASYNC(n) ((void)0)
#endif

typedef __attribute__((address_space(1))) v4i* gv4i_ptr;  // global 16B vector
typedef __attribute__((address_space(3))) v4i* lv4i_ptr;  // LDS 16B vector

namespace cfg {
constexpr int M  = 32;          // queries
constexpr int N  = 2048;        // model dim
constexpr int D  = 128;         // head dim
constexpr int L  = 32768;       // cache length
constexpr int H  = 16;          // heads
constexpr int TN = 3 * N;       // 6144, fused qkv width

constexpr int SPLITS = 32;      // L-splits per head -> 16*32 = 512 blocks
constexpr int SEG    = L / SPLITS;   // 1024 rows per split
constexpr int CH     = 128;     // K/V chunk rows staged per iteration
constexpr int KP     = 132;     // LDS pitch (dwords): %4==0, ==4 mod 64
constexpr int PP     = 132;     // LDS pitch for exp(S) tile
}

__device__ __forceinline__ v8f wmma4(v2f a, v2f b, v8f c) {
  // D = A(16x4,f32) x B(4x16,f32) + C(16x16,f32)
  return __builtin_amdgcn_wmma_f32_16x16x4_f32(false, a, false, b, (short)0, c,
                                               false, false);
}

// ---------------------------------------------------------------------------
// Kernel 1: inv_rms[row] = rsqrt(mean(X[row]^2))
// ---------------------------------------------------------------------------
__global__ __launch_bounds__(256) void k_rms(const float* __restrict__ X,
                                             float* __restrict__ invr) {
  const int row = blockIdx.x;
  const int t   = threadIdx.x;
  float s = 0.f;
  for (int i = t; i < cfg::N; i += 256) {
    float v = X[row * cfg::N + i];
    s += v * v;
  }
  __shared__ float red[256];
  red[t] = s;
  __syncthreads();
  for (int o = 128; o > 0; o >>= 1) {
    if (t < o) red[t] += red[t + o];
    __syncthreads();
  }
  if (t == 0) invr[row] = rsqrtf(red[0] / (float)cfg::N);
}

// ---------------------------------------------------------------------------
// Kernel 2: qkv = (X * invr) @ W_qkv   [32 x 6144]
// One wave per 16 output columns; both 16-row M-tiles per wave (B reused).
// ---------------------------------------------------------------------------
__global__ __launch_bounds__(128) void k_qkv(const float* __restrict__ X,
                                             const float* __restrict__ W,
                                             const float* __restrict__ invr,
                                             float* __restrict__ qkv) {
  const int wave = threadIdx.x >> 5;
  const int lane = threadIdx.x & 31;
  const int half = lane >> 4;
  const int lm   = lane & 15;
  const int n0   = (blockIdx.x * 4 + wave) * 16;

  const float r0 = invr[lm];
  const float r1 = invr[lm + 16];

  v8f c0 = {0.f, 0.f, 0.f, 0.f, 0.f, 0.f, 0.f, 0.f};
  v8f c1 = {0.f, 0.f, 0.f, 0.f, 0.f, 0.f, 0.f, 0.f};

  const float* xa = X + lm * cfg::N + 2 * half;
  const float* xb = X + (lm + 16) * cfg::N + 2 * half;
  const float* wb = W + (size_t)(2 * half) * cfg::TN + n0 + lm;

  for (int k0 = 0; k0 < cfg::N; k0 += 4) {
    v2f a0 = *(const v2f*)(xa + k0);
    v2f a1 = *(const v2f*)(xb + k0);
    a0 *= r0;
    a1 *= r1;
    v2f b;
    b.x = wb[(size_t)k0 * cfg::TN];
    b.y = wb[(size_t)(k0 + 1) * cfg::TN];
    c0 = wmma4(a0, b, c0);
    c1 = wmma4(a1, b, c1);
  }

#pragma unroll
  for (int v = 0; v < 8; ++v) {
    const int m0 = v + 8 * half;
    qkv[(size_t)m0 * cfg::TN + n0 + lm]        = c0[v];
    qkv[(size_t)(m0 + 16) * cfg::TN + n0 + lm] = c1[v];
  }
}

// ---------------------------------------------------------------------------
// K-chunk staging: issue per-thread copies of 128 rows x 128 f32 into an LDS
// buffer. Async path issues 16 GLOBAL_LOAD_ASYNC_TO_LDS_B128 per thread
// (tracked by ASYNCcnt, no VGPR data); fallback is float4 via registers.
// ---------------------------------------------------------------------------
__device__ __forceinline__ void stage_issue(const float* __restrict__ qkv,
                                            const float* __restrict__ cacheK,
                                            int h, int r0, int Ppos, int tid,
                                            float* __restrict__ buf) {
  using namespace cfg;
#pragma unroll
  for (int i = 0; i < (CH * D / 4) / 256; ++i) {   // 16 iterations
    const int e   = i * 256 + tid;                 // float4 index
    const int row = e >> 5;                        // 32 float4 per row
    const int c4  = e & 31;
    const int gr  = r0 + row;
    const unsigned nw = (unsigned)(gr - Ppos);
    const float* src = (nw < 32u)
                           ? (qkv + (size_t)nw * TN + N + h * D)  // fresh k
                           : (cacheK + ((size_t)h * L + gr) * D);
    src += c4 * 4;
    float* dst = buf + row * KP + c4 * 4;
#if USE_ASYNC_LDS
    float* srcnc = const_cast<float*>(src);
    __builtin_amdgcn_global_load_async_to_lds_b128((gv4i_ptr)srcnc,
                                                   (lv4i_ptr)dst, 0, 0);
#else
    *(float4*)dst = *(const float4*)src;
#endif
  }
}

// ---------------------------------------------------------------------------
// Kernel 3: streaming attention partials for one (head, L-split).
// 256 threads = 8 waves. Wave w owns K-rows [16w,16w+16) for scores and
// D-columns [16w,16w+16) for the output GEMM. K double-buffered in LDS.
// ---------------------------------------------------------------------------
__global__ __launch_bounds__(256) void k_attn(const float* __restrict__ qkv,
                                              const float* __restrict__ cacheK,
                                              const float* __restrict__ cacheV,
                                              const int* __restrict__ Pp,
                                              float* __restrict__ Opart,
                                              float* __restrict__ rspart) {
  using namespace cfg;
  __shared__ float k_lds[2 * CH * KP];  // 135,168 B double-buffered K chunk
  __shared__ float p_lds[M * PP];       //  16,896 B exp(S)
  __shared__ float rs_lds[M];

  const int split = blockIdx.x;
  const int h     = blockIdx.y;
  const int tid   = threadIdx.x;
  const int wave  = tid >> 5;
  const int lane  = tid & 31;
  const int half  = lane >> 4;
  const int lm    = lane & 15;
  const int Ppos  = *Pp;
  const int seg0  = split * SEG;

  if (tid < M) rs_lds[tid] = 0.f;

  const int wd0 = wave * 16;  // this wave's D-columns for the O GEMM
  v8f o0 = {0.f, 0.f, 0.f, 0.f, 0.f, 0.f, 0.f, 0.f};
  v8f o1 = {0.f, 0.f, 0.f, 0.f, 0.f, 0.f, 0.f, 0.f};
  float rs_acc = 0.f;  // thread t sums exp-row m=t>>3, cols [(t&7)*16, +16)

  const float* qbase = qkv + h * D;  // Q section, col = h*128 + d

  // prologue: start chunk 0 into buffer 0
  stage_issue(qkv, cacheK, h, seg0, Ppos, tid, &k_lds[0]);

  for (int c = 0; c < SEG; c += CH) {
    const int r0 = seg0 + c;
    const int cb = (c >> 7) & 1;
    float* curbuf = &k_lds[cb * CH * KP];

    // issue next chunk into the other buffer, then wait for current chunk.
    if (c + CH < SEG) {
      stage_issue(qkv, cacheK, h, r0 + CH, Ppos, tid,
                  &k_lds[(cb ^ 1) * CH * KP]);
      WAIT_ASYNC(16);  // async loads complete in order: current chunk done
    } else {
      WAIT_ASYNC(0);
    }
    __syncthreads();

    // ---- S = Q @ K^T for this wave's 16 K-rows ----
    const int wr = wave * 16;
    v8f s0 = {0.f, 0.f, 0.f, 0.f, 0.f, 0.f, 0.f, 0.f};
    v8f s1 = {0.f, 0.f, 0.f, 0.f, 0.f, 0.f, 0.f, 0.f};
    const float* krow = &curbuf[(wr + lm) * KP];
#pragma unroll 4
    for (int d0 = 0; d0 < D; d0 += 4) {
      const int dk = d0 + 2 * half;
      v2f a0 = *(const v2f*)(qbase + lm * TN + dk);
      v2f a1 = *(const v2f*)(qbase + (lm + 16) * TN + dk);
      v2f b;
      b.x = krow[dk];
      b.y = krow[dk + 1];
      s0 = wmma4(a0, b, s0);
      s1 = wmma4(a1, b, s1);
    }

    // ---- exp and publish P tile (32 x 16) ----
#pragma unroll
    for (int v = 0; v < 8; ++v) {
      const int m0 = v + 8 * half;
      p_lds[m0 * PP + wr + lm]        = __expf(s0[v]);
      p_lds[(m0 + 16) * PP + wr + lm] = __expf(s1[v]);
    }
    __syncthreads();

    // ---- rowsum partial ----
    {
      const int m  = tid >> 3;
      const int j0 = (tid & 7) * 16;
      float s = 0.f;
#pragma unroll
      for (int j = 0; j < 16; ++j) s += p_lds[m * PP + j0 + j];
      rs_acc += s;
    }

    // ---- O += P @ V_chunk, V streamed from HBM (coalesced per half-wave) ----
#pragma unroll 4
    for (int k0 = 0; k0 < CH; k0 += 4) {
      const int kk = k0 + 2 * half;
      v2f a0, a1;
      a0 = *(const v2f*)(&p_lds[lm * PP + kk]);
      a1 = *(const v2f*)(&p_lds[(lm + 16) * PP + kk]);
      const int rr = r0 + kk;
      const unsigned n0i = (unsigned)(rr - Ppos);
      const unsigned n1i = (unsigned)(rr + 1 - Ppos);
      const float* vb0 = (n0i < 32u)
                             ? (qkv + (size_t)n0i * TN + 2 * N + h * D)
                             : (cacheV + ((size_t)h * L + rr) * D);
      const float* vb1 = (n1i < 32u)
                             ? (qkv + (size_t)n1i * TN + 2 * N + h * D)
                             : (cacheV + ((size_t)h * L + rr + 1) * D);
      v2f b;
      b.x = vb0[wd0 + lm];
      b.y = vb1[wd0 + lm];
      o0 = wmma4(a0, b, o0);
      o1 = wmma4(a1, b, o1);
    }
    __syncthreads();
  }

  // ---- write partials ----
  atomicAdd(&rs_lds[tid >> 3], rs_acc);
  __syncthreads();

  float* ob = Opart + (size_t)(h * SPLITS + split) * M * D;
#pragma unroll
  for (int v = 0; v < 8; ++v) {
    const int m0 = v + 8 * half;
    ob[m0 * D + wd0 + lm]        = o0[v];
    ob[(m0 + 16) * D + wd0 + lm] = o1[v];
  }
  if (tid < M) rspart[(h * SPLITS + split) * M + tid] = rs_lds[tid];
}

// ---------------------------------------------------------------------------
// Kernel 4: combine split partials -> out[m][h*128+d]
// ---------------------------------------------------------------------------
__global__ __launch_bounds__(256) void k_reduce(const float* __restrict__ Opart,
                                                const float* __restrict__ rspart,
                                                float* __restrict__ out) {
  using namespace cfg;
  const int idx = blockIdx.x * 256 + threadIdx.x;  // 65536 total
  const int h = idx >> 12;
  const int m = (idx >> 7) & 31;
  const int d = idx & 127;
  float os = 0.f, rs = 0.f;
  for (int s = 0; s < SPLITS; ++s) {
    os += Opart[((size_t)(h * SPLITS + s) * M + m) * D + d];
    rs += rspart[(h * SPLITS + s) * M + m];
  }
  out[m * N + h * D + d] = os / rs;
}

// ---------------------------------------------------------------------------
extern "C" void kernel_launch(void* const* d_in, const int* in_sizes, int n_in,
                              void* d_out, int out_size, void* d_ws,
                              size_t ws_size, hipStream_t stream) {
  using namespace cfg;
  const float* X      = (const float*)d_in[0];
  const float* W_qkv  = (const float*)d_in[1];
  const float* cacheK = (const float*)d_in[2];
  const float* cacheV = (const float*)d_in[3];
  const int*   Pp     = (const int*)d_in[4];
  float* out = (float*)d_out;

  // workspace layout (bytes): [invr 128][pad to 256][qkv 786432][Opart 8MB][rs 64KB]
  char* ws = (char*)d_ws;
  float* invr   = (float*)ws;
  float* qkv    = (float*)(ws + 256);
  float* Opart  = (float*)(ws + 256 + (size_t)M * TN * 4);
  float* rspart = Opart + (size_t)H * SPLITS * M * D;

  k_rms<<<M, 256, 0, stream>>>(X, invr);
  k_qkv<<<(TN / 16) / 4, 128, 0, stream>>>(X, W_qkv, invr, qkv);
  k_attn<<<dim3(SPLITS, H), 256, 0, stream>>>(qkv, cacheK, cacheV, Pp, Opart,
                                              rspart);
  k_reduce<<<(H * M * D) / 256, 256, 0, stream>>>(Opart, rspart, out);
}